// ECGAttNet_31550829756492
// MI455X (gfx1250) — compile-verified
//
#include <hip/hip_runtime.h>
#include <math.h>

// ---------------- CDNA5 WMMA (gfx1250, wave32) ----------------
typedef float v2f __attribute__((ext_vector_type(2)));
typedef float v8f __attribute__((ext_vector_type(8)));

static __device__ __forceinline__ v8f wmma_f32_k4(v2f a, v2f b, v8f c) {
  // D = A(16x4,f32) * B(4x16,f32) + C(16x16,f32)
  return __builtin_amdgcn_wmma_f32_16x16x4_f32(
      /*neg_a=*/false, a, /*neg_b=*/false, b,
      /*c_mod=*/(short)0, c, /*reuse_a=*/false, /*reuse_b=*/false);
}

// ---------------- problem constants ----------------
// x:(4,256,64,64)  w:(512,256)  N=256 rows (b*64+h), M=N*64=16384 cols
#define CIN   256
#define OCH   512
#define NN    256      // B*H
#define WW    64
#define MM    16384    // NN*WW
#define GRPS  16
#define HC    8        // half group channels (q/k width)
#define EPSV  1e-5f

// ---------------- workspace layout (float offsets) ----------------
constexpr int QKV   = 0;                       // 512*16384
constexpr int EMB   = 8388608;                 // 32*64*64
constexpr int B1SC  = EMB + 131072;            // 512 bn1 scale
constexpr int B1SH  = B1SC + 512;              // 512 bn1 shift
constexpr int B2SUM = B1SH + 512;              // 48
constexpr int B2SQ  = B2SUM + 48;
constexpr int B2SC  = B2SQ + 48;
constexpr int B2SH  = B2SC + 48;
constexpr int B3SUM = B2SH + 48;               // 512
constexpr int B3SQ  = B3SUM + 512;
constexpr int B3SC  = B3SQ + 512;
constexpr int B3SH  = B3SC + 512;
constexpr int ORAW  = B3SH + 512;              // 256*512*64

// ---------------- helpers ----------------
static __device__ __forceinline__ float blockReduce256(float v, float* red, int t) {
  red[t] = v; __syncthreads();
  #pragma unroll
  for (int s = 128; s > 0; s >>= 1) {
    if (t < s) red[t] += red[t + s];
    __syncthreads();
  }
  float r = red[0]; __syncthreads();
  return r;
}

// ================= K0a: zero BN accumulators =================
__global__ void k_zero(float* ws) {
  int t = blockIdx.x * 256 + threadIdx.x;
  if (t < 96)        ws[B2SUM + t] = 0.0f;          // bn2 sum+sumsq
  else if (t < 1120) ws[B3SUM + (t - 96)] = 0.0f;   // bn3 sum+sumsq
}

// ================= K0b: emb[c][i][j] = rel_emb[c][i-j+63] =================
__global__ void k_emb(const float* __restrict__ rel, float* __restrict__ ws) {
  int gid = blockIdx.x * 256 + threadIdx.x;          // 32*64*64
  int c = gid >> 12, i = (gid >> 6) & 63, j = gid & 63;
  ws[EMB + gid] = rel[c * 127 + (i - j + 63)];
}

// ================= K1: qkv_raw(512x16384) = W(512x256) @ X(256x16384) =================
// one wave per block; wave computes a 16(o) x 64(m) strip with 4 fp32 WMMA accumulators.
// W tile staged into LDS via CDNA5 async global->LDS copy (ASYNCcnt path).
__global__ __launch_bounds__(32) void k_gemm_qkv(const float* __restrict__ x,
                                                 const float* __restrict__ w,
                                                 float* __restrict__ ws) {
  __shared__ float WT[16 * 256];                     // 16KB: W tile
  const int t   = threadIdx.x;                       // 0..31 (wave32)
  const int o0  = blockIdx.x * 16;
  const int m0  = blockIdx.y * 64;
  const int col = t & 15, hi = t >> 4;

  // stage 16x256 W tile into LDS: async b128 copies, one float4 per lane per issue
  #pragma unroll 4
  for (int it = 0; it < 32; ++it) {
    int idx = t + 32 * it;                           // 0..1023 float4 units
    int row = idx >> 6, cv = idx & 63;
    unsigned lds_off = (unsigned)(size_t)(&WT[row * 256 + cv * 4]);
    const float* gp  = w + (size_t)(o0 + row) * 256 + cv * 4;
    asm volatile("global_load_async_to_lds_b128 %0, %1, off"
                 :: "v"(lds_off), "v"(gp) : "memory");
  }
  asm volatile("s_wait_asynccnt 0x0" ::: "memory");
  __syncthreads();

  const float* xb = x + (size_t)(m0 >> 12) * (CIN * 4096) + (m0 & 4095);

  v8f acc0 = {}, acc1 = {}, acc2 = {}, acc3 = {};
  for (int k0 = 0; k0 < 256; k0 += 4) {
    v2f a;
    a.x = WT[col * 256 + k0 + 2 * hi + 0];
    a.y = WT[col * 256 + k0 + 2 * hi + 1];
    const float* xr0 = xb + (size_t)(k0 + 2 * hi) * 4096;
    if (k0 + 4 < 256) __builtin_prefetch(xb + (size_t)(k0 + 4) * 4096 + t * 8, 0, 0);
    v2f b;
    b.x = xr0[0 * 16 + col]; b.y = xr0[4096 + 0 * 16 + col];
    acc0 = wmma_f32_k4(a, b, acc0);
    b.x = xr0[1 * 16 + col]; b.y = xr0[4096 + 1 * 16 + col];
    acc1 = wmma_f32_k4(a, b, acc1);
    b.x = xr0[2 * 16 + col]; b.y = xr0[4096 + 2 * 16 + col];
    acc2 = wmma_f32_k4(a, b, acc2);
    b.x = xr0[3 * 16 + col]; b.y = xr0[4096 + 3 * 16 + col];
    acc3 = wmma_f32_k4(a, b, acc3);
  }

  float* q = ws + QKV;
  #pragma unroll
  for (int r = 0; r < 8; ++r) {
    int o = o0 + r + 8 * hi;
    q[(size_t)o * MM + m0 +  0 + col] = acc0[r];
    q[(size_t)o * MM + m0 + 16 + col] = acc1[r];
    q[(size_t)o * MM + m0 + 32 + col] = acc2[r];
    q[(size_t)o * MM + m0 + 48 + col] = acc3[r];
  }
}

// ================= K2: bn1 per-channel stats -> scale/shift =================
__global__ __launch_bounds__(256) void k_bn1(const float* __restrict__ gamma,
                                             const float* __restrict__ beta,
                                             float* __restrict__ ws) {
  __shared__ float red[256];
  const int o = blockIdx.x, t = threadIdx.x;
  const float* q = ws + QKV + (size_t)o * MM;
  float s = 0.f, sq = 0.f;
  for (int idx = t; idx < MM; idx += 256) { float v = q[idx]; s += v; sq += v * v; }
  s  = blockReduce256(s,  red, t);
  sq = blockReduce256(sq, red, t);
  if (t == 0) {
    float mean = s / (float)MM;
    float var  = sq / (float)MM - mean * mean;
    float sc   = gamma[o] * rsqrtf(var + EPSV);
    ws[B1SC + o] = sc;
    ws[B1SH + o] = beta[o] - mean * sc;
  }
}

// ================= K3: bn2 stats (qk/qe/ke moments), per (n,g) block =================
__global__ __launch_bounds__(256) void k_bn2_stats(float* __restrict__ ws) {
  __shared__ float buf[512 + 512 + 4096 + 4096 + 256];
  float* sQ   = buf;
  float* sK   = buf + 512;
  float* sQET = buf + 1024;          // QET[b*64+a] = qe[a][b]
  float* sKE  = buf + 5120;          // KE [a*64+b] = ke[a][b]
  float* red  = buf + 9216;

  const int t = threadIdx.x;
  const int n = blockIdx.x >> 4, g = blockIdx.x & 15;
  const float* qkv = ws + QKV;
  const float* emb = ws + EMB;

  // load q,k with bn1 applied
  for (int idx = t; idx < 1024; idx += 256) {
    int c2 = idx >> 6, i = idx & 63;
    int o = g * 32 + c2;
    float v = qkv[(size_t)o * MM + n * 64 + i] * ws[B1SC + o] + ws[B1SH + o];
    if (c2 < HC) sQ[c2 * 64 + i] = v; else sK[(c2 - HC) * 64 + i] = v;
  }
  __syncthreads();

  // qe[a][b], ke[a][b] : rank-8 contractions against emb
  for (int idx = t; idx < 4096; idx += 256) {
    int a = idx >> 6, b = idx & 63;
    float vq = 0.f, vk = 0.f;
    #pragma unroll
    for (int c = 0; c < HC; ++c) {
      float e_q = emb[c * 4096 + a * 64 + b];
      float e_k = emb[(HC + c) * 4096 + a * 64 + b];
      vq += sQ[c * 64 + a] * e_q;
      vk += sK[c * 64 + a] * e_k;
    }
    sQET[b * 64 + a] = vq;
    sKE [a * 64 + b] = vk;
  }
  __syncthreads();

  // qk = qe^T @ ke via fp32 WMMA; 16 (i,j) tiles over 8 waves
  const int wv = t >> 5, lane = t & 31, col = lane & 15, hi = lane >> 4;
  float pqk_s = 0.f, pqk_q = 0.f;
  #pragma unroll
  for (int tt = wv * 2; tt < wv * 2 + 2; ++tt) {
    int i0 = (tt >> 2) * 16, j0 = (tt & 3) * 16;
    v8f acc = {};
    for (int t0 = 0; t0 < 64; t0 += 4) {
      v2f a, b;
      a.x = sQET[(i0 + col) * 64 + t0 + 2 * hi + 0];
      a.y = sQET[(i0 + col) * 64 + t0 + 2 * hi + 1];
      b.x = sKE[(t0 + 2 * hi + 0) * 64 + j0 + col];
      b.y = sKE[(t0 + 2 * hi + 1) * 64 + j0 + col];
      acc = wmma_f32_k4(a, b, acc);
    }
    #pragma unroll
    for (int r = 0; r < 8; ++r) { float v = acc[r]; pqk_s += v; pqk_q += v * v; }
  }
  // qe/ke moments
  float pqe_s = 0.f, pqe_q = 0.f, pke_s = 0.f, pke_q = 0.f;
  for (int idx = t; idx < 4096; idx += 256) {
    float vq = sQET[idx], vk = sKE[idx];
    pqe_s += vq; pqe_q += vq * vq; pke_s += vk; pke_q += vk * vk;
  }
  float r0 = blockReduce256(pqk_s, red, t);
  float r1 = blockReduce256(pqk_q, red, t);
  float r2 = blockReduce256(pqe_s, red, t);
  float r3 = blockReduce256(pqe_q, red, t);
  float r4 = blockReduce256(pke_s, red, t);
  float r5 = blockReduce256(pke_q, red, t);
  if (t == 0) {
    atomicAdd(ws + B2SUM + g,        r0); atomicAdd(ws + B2SQ + g,        r1);
    atomicAdd(ws + B2SUM + 16 + g,   r2); atomicAdd(ws + B2SQ + 16 + g,   r3);
    atomicAdd(ws + B2SUM + 32 + g,   r4); atomicAdd(ws + B2SQ + 32 + g,   r5);
  }
}

// ================= K4: finalize bn2 =================
__global__ void k_bn2_fin(const float* __restrict__ gamma,
                          const float* __restrict__ beta,
                          float* __restrict__ ws) {
  int c = threadIdx.x;
  if (c < 48) {
    const float cnt = (float)NN * 4096.0f;
    float mean = ws[B2SUM + c] / cnt;
    float var  = ws[B2SQ + c] / cnt - mean * mean;
    float sc   = gamma[c] * rsqrtf(var + EPSV);
    ws[B2SC + c] = sc;
    ws[B2SH + c] = beta[c] - mean * sc;
  }
}

// ================= K5: fused attention apply per (n,g) =================
__global__ __launch_bounds__(256) void k_attn(float* __restrict__ ws) {
  __shared__ float buf[14592];
  float* sQ   = buf;                  // 512
  float* sK   = buf + 512;            // 512
  float* sVT  = buf + 1024;           // 1024 : VT[j*16+c]
  float* sQET = buf + 2048;           // 4096 (aliased later by sAO/sAE)
  float* sKE  = buf + 6144;           // 4096
  float* sSIM = buf + 10240;          // 4096
  float* red  = buf + 14336;          // 256
  float* sAO  = buf + 2048;           // 1024 (phase-5 alias)
  float* sAE  = buf + 3072;           // 1024 (phase-5 alias)

  const int t = threadIdx.x;
  const int n = blockIdx.x >> 4, g = blockIdx.x & 15;
  const float* qkv = ws + QKV;
  const float* emb = ws + EMB;

  // phase 1: load q,k,v with bn1 applied
  for (int idx = t; idx < 1024; idx += 256) {
    int c2 = idx >> 6, i = idx & 63;
    int o = g * 32 + c2;
    float v = qkv[(size_t)o * MM + n * 64 + i] * ws[B1SC + o] + ws[B1SH + o];
    if (c2 < HC) sQ[c2 * 64 + i] = v; else sK[(c2 - HC) * 64 + i] = v;
  }
  for (int idx = t; idx < 1024; idx += 256) {
    int c2 = idx >> 6, j = idx & 63;
    int o = g * 32 + 16 + c2;
    sVT[j * 16 + c2] = qkv[(size_t)o * MM + n * 64 + j] * ws[B1SC + o] + ws[B1SH + o];
  }
  __syncthreads();

  // phase 2: recompute qe (transposed) and ke
  for (int idx = t; idx < 4096; idx += 256) {
    int a = idx >> 6, b = idx & 63;
    float vq = 0.f, vk = 0.f;
    #pragma unroll
    for (int c = 0; c < HC; ++c) {
      vq += sQ[c * 64 + a] * emb[c * 4096 + a * 64 + b];
      vk += sK[c * 64 + a] * emb[(HC + c) * 4096 + a * 64 + b];
    }
    sQET[b * 64 + a] = vq;
    sKE [a * 64 + b] = vk;
  }
  __syncthreads();

  // phase 3: qk WMMA + bn2-affine combine -> SIM
  const int wv = t >> 5, lane = t & 31, col = lane & 15, hi = lane >> 4;
  const float s0 = ws[B2SC + g],      zall = ws[B2SH + g] + ws[B2SH + 16 + g] + ws[B2SH + 32 + g];
  const float s1 = ws[B2SC + 16 + g], s2 = ws[B2SC + 32 + g];
  #pragma unroll
  for (int tt = wv * 2; tt < wv * 2 + 2; ++tt) {
    int i0 = (tt >> 2) * 16, j0 = (tt & 3) * 16;
    v8f acc = {};
    for (int t0 = 0; t0 < 64; t0 += 4) {
      v2f a, b;
      a.x = sQET[(i0 + col) * 64 + t0 + 2 * hi + 0];
      a.y = sQET[(i0 + col) * 64 + t0 + 2 * hi + 1];
      b.x = sKE[(t0 + 2 * hi + 0) * 64 + j0 + col];
      b.y = sKE[(t0 + 2 * hi + 1) * 64 + j0 + col];
      acc = wmma_f32_k4(a, b, acc);
    }
    #pragma unroll
    for (int r = 0; r < 8; ++r) {
      int i = i0 + r + 8 * hi, j = j0 + col;
      sSIM[i * 64 + j] = s0 * acc[r] + zall
                       + s1 * sQET[j * 64 + i]     // qe[i][j]
                       + s2 * sKE [i * 64 + j];    // ke[i][j]
    }
  }
  __syncthreads();

  // phase 4: softmax over j (row per thread, t<64)
  if (t < 64) {
    float* row = sSIM + t * 64;
    float m = row[0];
    for (int j = 1; j < 64; ++j) m = fmaxf(m, row[j]);
    float s = 0.f;
    for (int j = 0; j < 64; ++j) { float e = __expf(row[j] - m); row[j] = e; s += e; }
    float inv = 1.0f / s;
    for (int j = 0; j < 64; ++j) row[j] *= inv;
  }
  __syncthreads();

  // phase 5: attn (WMMA, waves 0-3) + attn_e (FMA, waves 4-7)
  if (wv < 4) {
    int i0 = wv * 16;
    v8f acc = {};
    for (int j0 = 0; j0 < 64; j0 += 4) {
      v2f a, b;
      a.x = sSIM[(i0 + col) * 64 + j0 + 2 * hi + 0];
      a.y = sSIM[(i0 + col) * 64 + j0 + 2 * hi + 1];
      b.x = sVT[(j0 + 2 * hi + 0) * 16 + col];
      b.y = sVT[(j0 + 2 * hi + 1) * 16 + col];
      acc = wmma_f32_k4(a, b, acc);
    }
    #pragma unroll
    for (int r = 0; r < 8; ++r) sAO[col * 64 + i0 + r + 8 * hi] = acc[r];
  } else {
    int t2 = t - 128;
    for (int rep = 0; rep < 8; ++rep) {
      int idx = rep * 128 + t2;
      int c2 = idx >> 6, i = idx & 63;
      const float* ev = emb + (size_t)(16 + c2) * 4096 + i * 64;
      const float* sr = sSIM + i * 64;
      float s = 0.f;
      #pragma unroll 8
      for (int j = 0; j < 64; ++j) s += sr[j] * ev[j];
      sAE[c2 * 64 + i] = s;
    }
  }
  __syncthreads();

  // phase 6: store out_raw + bn3 moment accumulation
  // thread t: local channel c2 = t>>3 (0..31), covers 8 i's
  {
    int c2 = t >> 3, l8 = t & 7;
    const float* src = (c2 < 16) ? (sAO + c2 * 64) : (sAE + (c2 - 16) * 64);
    int ch = (c2 < 16) ? (g * 32 + 2 * c2) : (g * 32 + 2 * (c2 - 16) + 1);
    float* dst = ws + ORAW + ((size_t)n * 512 + ch) * 64;
    float ps = 0.f, pq = 0.f;
    #pragma unroll
    for (int k = 0; k < 8; ++k) {
      int i = l8 * 8 + k;
      float v = src[i];
      dst[i] = v;
      ps += v; pq += v * v;
    }
    red[t] = ps; __syncthreads();
    if (l8 == 0) {
      float s = 0.f;
      #pragma unroll
      for (int k = 0; k < 8; ++k) s += red[t + k];
      atomicAdd(ws + B3SUM + ch, s);
    }
    __syncthreads();
    red[t] = pq; __syncthreads();
    if (l8 == 0) {
      float s = 0.f;
      #pragma unroll
      for (int k = 0; k < 8; ++k) s += red[t + k];
      atomicAdd(ws + B3SQ + ch, s);
    }
  }
}

// ================= K6: finalize bn3 =================
__global__ void k_bn3_fin(const float* __restrict__ gamma,
                          const float* __restrict__ beta,
                          float* __restrict__ ws) {
  int c = blockIdx.x * 256 + threadIdx.x;
  if (c < 512) {
    const float cnt = (float)(NN * WW);
    float mean = ws[B3SUM + c] / cnt;
    float var  = ws[B3SQ + c] / cnt - mean * mean;
    float sc   = gamma[c] * rsqrtf(var + EPSV);
    ws[B3SC + c] = sc;
    ws[B3SH + c] = beta[c] - mean * sc;
  }
}

// ================= K7: bn3 apply + pair sum + transpose -> y(4,256,64,64) =================
__global__ __launch_bounds__(256) void k_final(const float* __restrict__ ws,
                                               float* __restrict__ out) {
  int gid = blockIdx.x * 256 + threadIdx.x;          // 4*256*64*64
  int i  = gid & 63;
  int h  = (gid >> 6) & 63;
  int oc = (gid >> 12) & 255;
  int b  = gid >> 20;
  int n  = b * 64 + h;
  int c0 = 2 * oc, c1 = 2 * oc + 1;
  const float* o0 = ws + ORAW + ((size_t)n * 512 + c0) * 64;
  float v = o0[i]      * ws[B3SC + c0] + ws[B3SH + c0]
          + o0[64 + i] * ws[B3SC + c1] + ws[B3SH + c1];
  out[gid] = v;
}

// ================= launch =================
extern "C" void kernel_launch(void* const* d_in, const int* in_sizes, int n_in,
                              void* d_out, int out_size, void* d_ws, size_t ws_size,
                              hipStream_t stream) {
  const float* x       = (const float*)d_in[0];
  const float* w_qkv   = (const float*)d_in[1];
  const float* bn1_g   = (const float*)d_in[2];
  const float* bn1_b   = (const float*)d_in[3];
  const float* rel_emb = (const float*)d_in[4];
  const float* bn2_g   = (const float*)d_in[5];
  const float* bn2_b   = (const float*)d_in[6];
  const float* bn3_g   = (const float*)d_in[7];
  const float* bn3_b   = (const float*)d_in[8];
  float* ws  = (float*)d_ws;
  float* out = (float*)d_out;

  k_zero<<<5, 256, 0, stream>>>(ws);
  k_emb<<<512, 256, 0, stream>>>(rel_emb, ws);
  k_gemm_qkv<<<dim3(32, 256), 32, 0, stream>>>(x, w_qkv, ws);
  k_bn1<<<512, 256, 0, stream>>>(bn1_g, bn1_b, ws);
  k_bn2_stats<<<4096, 256, 0, stream>>>(ws);
  k_bn2_fin<<<1, 64, 0, stream>>>(bn2_g, bn2_b, ws);
  k_attn<<<4096, 256, 0, stream>>>(ws);
  k_bn3_fin<<<2, 256, 0, stream>>>(bn3_g, bn3_b, ws);
  k_final<<<16384, 256, 0, stream>>>(ws, out);
}